// WassersteinIB_43224550867323
// MI455X (gfx1250) — compile-verified
//
#include <hip/hip_runtime.h>
#include <hip/hip_bf16.h>
#include <math.h>

// ---------------------------------------------------------------------------
// WMMA f32 16x16x4 helpers (CDNA5 / gfx1250, wave32)
// A fragment (16x4 f32): 2 VGPRs. lanes 0-15: M=lane, VGPR0=K0, VGPR1=K1
//                                 lanes16-31: M=lane-16, VGPR0=K2, VGPR1=K3
// B fragment (4x16 f32): 2 VGPRs. lanes 0-15: N=lane, VGPR0=K0, VGPR1=K1
//                                 lanes16-31: N=lane-16, VGPR0=K2, VGPR1=K3
// C/D (16x16 f32): 8 VGPRs. lane<16: N=lane, M=vgpr;  lane>=16: N=lane-16, M=vgpr+8
// ---------------------------------------------------------------------------
typedef float v2f __attribute__((ext_vector_type(2)));
typedef float v8f __attribute__((ext_vector_type(8)));

static __device__ inline v8f wmma_f32(v2f a, v2f b, v8f c) {
  return __builtin_amdgcn_wmma_f32_16x16x4_f32(
      /*neg_a=*/false, a, /*neg_b=*/false, b,
      /*c_mod=*/(short)0, c, /*reuse_a=*/false, /*reuse_b=*/false);
}

// Async DMA: 16B global -> LDS, tracked on ASYNCcnt (CDNA5).
// GVS addressing: mem = SGPR64 base + 32-bit unsigned VGPR byte offset.
static __device__ inline void async_load_b128(unsigned lds_addr, unsigned voff,
                                              unsigned long long saddr) {
  asm volatile("global_load_async_to_lds_b128 %0, %1, %2"
               :: "v"(lds_addr), "v"(voff), "s"(saddr)
               : "memory");
}
static __device__ inline void wait_async_9() {
  asm volatile("s_wait_asynccnt 0x9" ::: "memory");
}
static __device__ inline void wait_async_0() {
  asm volatile("s_wait_asynccnt 0x0" ::: "memory");
}

// Load A fragment from row-major A (lda floats per row), tile at (mrow0, k0)
static __device__ inline v2f load_a_frag(const float* __restrict__ A, int lda,
                                         int mrow0, int k0, int lane) {
  int l = lane & 15;
  int koff = (lane >> 4) << 1;
  v2f a;
  a.x = A[(size_t)(mrow0 + l) * lda + k0 + koff];
  a.y = A[(size_t)(mrow0 + l) * lda + k0 + koff + 1];
  return a;
}

// Load B fragment from row-major B (K x N, ldb floats per row), tile at (k0, n0)
static __device__ inline v2f load_b_frag(const float* __restrict__ B, int ldb,
                                         int k0, int n0, int lane) {
  int l = lane & 15;
  int koff = (lane >> 4) << 1;
  v2f b;
  b.x = B[(size_t)(k0 + koff) * ldb + n0 + l];
  b.y = B[(size_t)(k0 + koff + 1) * ldb + n0 + l];
  return b;
}

static __device__ inline float gelu_exact(float x) {
  return 0.5f * x * (1.0f + erff(x * 0.70710678118654752f));
}
static __device__ inline float symlogf(float t) {
  return copysignf(log1pf(fabsf(t)), t);
}

#define B_SZ 256
#define DIN 65536
#define DZ 64
#define DOUT 65536
#define H_SZ 128
#define SPLITK 64   // enc1: K-slab = 65536/64 = 1024
#define KCHUNK 64   // enc1: async-staged K chunk

// LDS row strides chosen so the two half-wave K-rows of a B fragment land on
// disjoint bank ranges (stride mod 32 floats == 8 -> 2*stride mod 64 == 16).
#define A_STRIDE 68
#define B_STRIDE 136

// ---------------------------------------------------------------------------
// 0) init: zero split-K accumulator + loss scalars
// ---------------------------------------------------------------------------
__global__ void init_kernel(float* h1, float* recon, float* Wacc) {
  int idx = blockIdx.x * blockDim.x + threadIdx.x;
  if (idx < B_SZ * H_SZ) h1[idx] = 0.0f;
  if (idx == 0) { recon[0] = 0.0f; Wacc[0] = 0.0f; }
}

// ---------------------------------------------------------------------------
// 1) enc1: h1 += x[256x65536] @ enc_w1[65536x128], split-K w/ f32 atomics.
// grid = (SPLITK, 16 Mtiles), block = 256 (8 waves = 8 Ntiles of 16).
// A (16 x 64) and B (64 x 128) chunks double-buffered through LDS via
// GLOBAL_LOAD_ASYNC_TO_LDS_B128; s_wait_asynccnt overlaps DMA with WMMA.
// ---------------------------------------------------------------------------
// Issue one chunk: 1 A-load + 8 B-loads of 16B per thread (ASYNCcnt += 9).
static __device__ inline void enc1_stage(float (*As)[A_STRIDE],
                                         float (*Bs)[B_STRIDE],
                                         const float* x, const float* w1,
                                         int tid, int mrow0, int kchunk) {
  unsigned long long xs = (unsigned long long)(uintptr_t)x;
  unsigned long long w1s = (unsigned long long)(uintptr_t)w1;
  // A: 16 rows x 64 floats (256B/row = 16 lanes * 16B)
  int ar = tid >> 4;
  int ac = tid & 15;
  async_load_b128((unsigned)(uintptr_t)&As[ar][ac * 4],
                  (unsigned)((mrow0 + ar) * DIN + kchunk) * 4u + (unsigned)(ac * 16),
                  xs);
  // B: 64 rows x 128 floats (512B/row = 32 * 16B), 8 per thread
#pragma unroll
  for (int j = 0; j < 8; ++j) {
    int i = tid + j * 256;
    int r = i >> 5;
    int cb = i & 31;
    async_load_b128((unsigned)(uintptr_t)&Bs[r][cb * 4],
                    (unsigned)(kchunk + r) * (H_SZ * 4u) + (unsigned)(cb * 16),
                    w1s);
  }
}

static __device__ inline v8f enc1_consume(const float (*As)[A_STRIDE],
                                          const float (*Bs)[B_STRIDE],
                                          int lane, int n0, v8f acc) {
  int l = lane & 15;
  int koff = (lane >> 4) << 1;
#pragma unroll
  for (int kk = 0; kk < KCHUNK; kk += 4) {
    v2f a, b;
    a.x = As[l][kk + koff];
    a.y = As[l][kk + koff + 1];
    b.x = Bs[kk + koff][n0 + l];
    b.y = Bs[kk + koff + 1][n0 + l];
    acc = wmma_f32(a, b, acc);
  }
  return acc;
}

__global__ void __launch_bounds__(256) enc1_kernel(const float* __restrict__ x,
                                                   const float* __restrict__ w1,
                                                   float* __restrict__ h1) {
  __shared__ float As[2][16][A_STRIDE];
  __shared__ float Bs[2][KCHUNK][B_STRIDE];
  int tid = threadIdx.x;
  int lane = tid & 31;
  int wid = tid >> 5;
  int mrow0 = blockIdx.y * 16;
  int n0 = wid * 16;
  int k0 = blockIdx.x * (DIN / SPLITK);
  const int NCHUNK = (DIN / SPLITK) / KCHUNK;  // 16

  enc1_stage(As[0], Bs[0], x, w1, tid, mrow0, k0);
  v8f acc = {};
  for (int c = 0; c < NCHUNK; ++c) {
    int bsel = c & 1;
    if (c + 1 < NCHUNK) {
      enc1_stage(As[bsel ^ 1], Bs[bsel ^ 1], x, w1, tid, mrow0,
                 k0 + (c + 1) * KCHUNK);
      wait_async_9();  // chunk c landed; chunk c+1 still in flight
    } else {
      wait_async_0();
    }
    __syncthreads();
    acc = enc1_consume(As[bsel], Bs[bsel], lane, n0, acc);
    __syncthreads();  // buffer reuse safety
  }
  int col = n0 + (lane & 15);
  int mbase = mrow0 + ((lane >> 4) << 3);
#pragma unroll
  for (int i = 0; i < 8; ++i)
    atomicAdd(&h1[(mbase + i) * H_SZ + col], acc[i]);
}

// ---------------------------------------------------------------------------
// 2) bias + exact GELU elementwise over 256x128
// ---------------------------------------------------------------------------
__global__ void biasgelu_kernel(const float* __restrict__ h1,
                                const float* __restrict__ b1,
                                float* __restrict__ h1g) {
  int idx = blockIdx.x * blockDim.x + threadIdx.x;  // 32768
  h1g[idx] = gelu_exact(h1[idx] + b1[idx & (H_SZ - 1)]);
}

// ---------------------------------------------------------------------------
// 3) small GEMM (M=256, N=128, K=KDIM) + bias (+ optional GELU)
// grid = 16 Mtiles, block = 256 (8 waves = 8 Ntiles)
// ---------------------------------------------------------------------------
template <int KDIM, int LDA, bool DO_GELU>
__global__ void small_gemm_kernel(const float* __restrict__ A,
                                  const float* __restrict__ Bm,
                                  const float* __restrict__ bias,
                                  float* __restrict__ out) {
  int tid = threadIdx.x;
  int lane = tid & 31;
  int wid = tid >> 5;
  int mrow0 = blockIdx.x * 16;
  int n0 = wid * 16;
  v8f acc = {};
  for (int k = 0; k < KDIM; k += 4) {
    v2f a = load_a_frag(A, LDA, mrow0, k, lane);
    v2f b = load_b_frag(Bm, H_SZ, k, n0, lane);
    acc = wmma_f32(a, b, acc);
  }
  int col = n0 + (lane & 15);
  int mbase = mrow0 + ((lane >> 4) << 3);
  float bv = bias[col];
#pragma unroll
  for (int i = 0; i < 8; ++i) {
    float v = acc[i] + bv;
    if (DO_GELU) v = gelu_exact(v);
    out[(mbase + i) * H_SZ + col] = v;
  }
}

// ---------------------------------------------------------------------------
// 4) reparameterize: z = mu + exp(0.5*logvar)*eps  (256x64)
// ---------------------------------------------------------------------------
__global__ void z_kernel(const float* __restrict__ h,
                         const float* __restrict__ eps,
                         float* __restrict__ z) {
  int idx = blockIdx.x * blockDim.x + threadIdx.x;  // 16384
  int r = idx >> 6, c = idx & 63;
  float mu = h[r * H_SZ + c];
  float logvar = h[r * H_SZ + DZ + c];
  z[idx] = mu + expf(0.5f * logvar) * eps[idx];
}

// ---------------------------------------------------------------------------
// 5) fused dec2 + symlog recon loss. grid = 512 (N blocks of 128 cols),
// block = 256 (8 waves = 8 Ntiles). The 128x128 dec_w2 block is async-DMA'd
// into LDS ONCE per workgroup (instead of re-read 16x from global), then 16
// Mtiles x K=128 WMMA with B from LDS. y_pred never touches memory.
// ---------------------------------------------------------------------------
__global__ void __launch_bounds__(256)
dec2_recon_kernel(const float* __restrict__ g, const float* __restrict__ w2,
                  const float* __restrict__ b2, const float* __restrict__ y,
                  float* __restrict__ recon_acc) {
  __shared__ float Bs[H_SZ][B_STRIDE];
  int tid = threadIdx.x;
  int lane = tid & 31;
  int wid = tid >> 5;
  int nblk = blockIdx.x * 128;
  int n0 = wid * 16;  // within block

  // Stage w2[0:128][nblk:nblk+128]: 128 rows x 512B (32 x 16B per row),
  // 16 async b128 loads per thread.
  unsigned long long w2s = (unsigned long long)(uintptr_t)w2;
#pragma unroll
  for (int j = 0; j < 16; ++j) {
    int i = tid + j * 256;  // 0..4095
    int r = i >> 5;
    int cb = i & 31;
    async_load_b128((unsigned)(uintptr_t)&Bs[r][cb * 4],
                    (unsigned)r * (DOUT * 4u) + (unsigned)(nblk * 4) +
                        (unsigned)(cb * 16),
                    w2s);
  }
  wait_async_0();
  __syncthreads();

  int l = lane & 15;
  int koff = (lane >> 4) << 1;
  float lsum = 0.0f;
  for (int mt = 0; mt < 16; ++mt) {
    int mrow0 = mt * 16;
    // prefetch this tile's y rows into cache (global_prefetch_b8)
    __builtin_prefetch(&y[(size_t)(mrow0 + l) * DOUT + nblk + n0], 0, 1);
    v8f acc = {};
#pragma unroll
    for (int k = 0; k < H_SZ; k += 4) {
      v2f a = load_a_frag(g, H_SZ, mrow0, k, lane);
      v2f b;
      b.x = Bs[k + koff][n0 + l];
      b.y = Bs[k + koff + 1][n0 + l];
      acc = wmma_f32(a, b, acc);
    }
    int col = nblk + n0 + l;
    int mbase = mrow0 + ((lane >> 4) << 3);
    float bias = b2[col];
#pragma unroll
    for (int i = 0; i < 8; ++i) {
      int m = mbase + i;
      float yp = acc[i] + bias;
      float d = symlogf(yp) - symlogf(y[(size_t)m * DOUT + col]);
      lsum += d * d;
    }
  }
  __shared__ float red[256];
  red[tid] = lsum;
  __syncthreads();
  for (int off = 128; off > 0; off >>= 1) {
    if (tid < off) red[tid] += red[tid + off];
    __syncthreads();
  }
  if (tid == 0) atomicAdd(recon_acc, red[0]);
}

// ---------------------------------------------------------------------------
// 6) row squared norms of z and prior (each 256x64)
// ---------------------------------------------------------------------------
__global__ void rowsq_kernel(const float* __restrict__ z,
                             const float* __restrict__ prior,
                             float* __restrict__ x2, float* __restrict__ y2) {
  int i = blockIdx.x * blockDim.x + threadIdx.x;  // 512
  if (i < B_SZ) {
    float s = 0.0f;
    for (int k = 0; k < DZ; ++k) { float t = z[i * DZ + k]; s += t * t; }
    x2[i] = s;
  } else if (i < 2 * B_SZ) {
    int j = i - B_SZ;
    float s = 0.0f;
    for (int k = 0; k < DZ; ++k) { float t = prior[j * DZ + k]; s += t * t; }
    y2[j] = s;
  }
}

// ---------------------------------------------------------------------------
// 7) C = clip(x2+y2-2 z@prior^T, 0), K = exp(-C/eps). WMMA for z@prior^T.
// grid = 32, block = 256: 8 tiles per wg, 256 16x16 tiles total.
// ---------------------------------------------------------------------------
__global__ void ck_kernel(const float* __restrict__ z,
                          const float* __restrict__ prior,
                          const float* __restrict__ x2,
                          const float* __restrict__ y2,
                          float* __restrict__ Cmat, float* __restrict__ Kmat) {
  int tid = threadIdx.x;
  int lane = tid & 31;
  int wid = tid >> 5;
  int t = blockIdx.x * 8 + wid;
  int mrow0 = (t >> 4) * 16;
  int n0 = (t & 15) * 16;
  v8f acc = {};
  int l = lane & 15;
  int koff = (lane >> 4) << 1;
  for (int k = 0; k < DZ; k += 4) {
    v2f a = load_a_frag(z, DZ, mrow0, k, lane);
    v2f b;  // B = prior^T : B[k][n] = prior[n*DZ + k]
    b.x = prior[(n0 + l) * DZ + k + koff];
    b.y = prior[(n0 + l) * DZ + k + koff + 1];
    acc = wmma_f32(a, b, acc);
  }
  int col = n0 + l;
  int mbase = mrow0 + ((lane >> 4) << 3);
#pragma unroll
  for (int i = 0; i < 8; ++i) {
    int m = mbase + i;
    float c = fmaxf(x2[m] + y2[col] - 2.0f * acc[i], 0.0f);
    Cmat[m * B_SZ + col] = c;
    Kmat[m * B_SZ + col] = expf(-c * 10.0f);  // 1/SINK_EPS = 10
  }
}

// ---------------------------------------------------------------------------
// 8) Sinkhorn iterations: single workgroup, 256 threads, 50 iterations with
// exact done|conv masking semantics from the reference.
// ---------------------------------------------------------------------------
__global__ void sinkhorn_iter_kernel(const float* __restrict__ Kmat,
                                     float* __restrict__ uG,
                                     float* __restrict__ vG) {
  __shared__ float u[B_SZ], v[B_SZ], vn[B_SZ], red[B_SZ];
  __shared__ int done;
  int tid = threadIdx.x;
  u[tid] = 1.0f;
  v[tid] = 1.0f;
  if (tid == 0) done = 0;
  __syncthreads();
  const float mass = 1.0f / 256.0f;  // mu = nu = 1/n
  for (int it = 0; it < 50; ++it) {
    // v_new = nu / (K^T u + 1e-8): column `tid` dot u
    float s = 0.0f;
    for (int r = 0; r < B_SZ; ++r) s += Kmat[r * B_SZ + tid] * u[r];
    float vnew = mass / (s + 1e-8f);
    vn[tid] = vnew;
    __syncthreads();
    // u_cand = mu / (K v_new + 1e-8): row `tid` dot v_new
    float s2 = 0.0f;
    for (int c = 0; c < B_SZ; ++c) s2 += Kmat[tid * B_SZ + c] * vn[c];
    float ucand = mass / (s2 + 1e-8f);
    red[tid] = fabsf(ucand - u[tid]);
    __syncthreads();
    for (int off = 128; off > 0; off >>= 1) {
      if (tid < off) red[tid] = fmaxf(red[tid], red[tid + off]);
      __syncthreads();
    }
    bool conv = red[0] < 1e-3f;
    int d0 = done;
    __syncthreads();
    if (!(d0 || conv)) u[tid] = ucand;   // u frozen once done|conv
    if (!d0) v[tid] = vn[tid];           // v frozen once done
    if (tid == 0 && conv) done = 1;
    __syncthreads();
  }
  uG[tid] = u[tid];
  vG[tid] = v[tid];
}

// ---------------------------------------------------------------------------
// 9) W = sum_ij u_i K_ij v_j C_ij
// ---------------------------------------------------------------------------
__global__ void wsum_kernel(const float* __restrict__ Kmat,
                            const float* __restrict__ Cmat,
                            const float* __restrict__ uG,
                            const float* __restrict__ vG,
                            float* __restrict__ Wacc) {
  int tid = threadIdx.x;
  float s = 0.0f;
  for (int idx = blockIdx.x * blockDim.x + tid; idx < B_SZ * B_SZ;
       idx += gridDim.x * blockDim.x) {
    int r = idx >> 8, c = idx & 255;
    s += uG[r] * Kmat[idx] * vG[c] * Cmat[idx];
  }
  __shared__ float red[256];
  red[tid] = s;
  __syncthreads();
  for (int off = 128; off > 0; off >>= 1) {
    if (tid < off) red[tid] += red[tid + off];
    __syncthreads();
  }
  if (tid == 0) atomicAdd(Wacc, red[0]);
}

// ---------------------------------------------------------------------------
// 10) finalize scalar
// ---------------------------------------------------------------------------
__global__ void finalize_kernel(const float* __restrict__ recon,
                                const float* __restrict__ Wacc,
                                float* __restrict__ out) {
  out[0] = recon[0] * (1.0f / (256.0f * 65536.0f)) + 0.1f * Wacc[0];
}

// ---------------------------------------------------------------------------
extern "C" void kernel_launch(void* const* d_in, const int* in_sizes, int n_in,
                              void* d_out, int out_size, void* d_ws, size_t ws_size,
                              hipStream_t stream) {
  const float* x       = (const float*)d_in[0];
  const float* y       = (const float*)d_in[1];
  const float* eps     = (const float*)d_in[2];
  const float* enc_w1  = (const float*)d_in[3];
  const float* enc_b1  = (const float*)d_in[4];
  const float* enc_w2  = (const float*)d_in[5];
  const float* enc_b2  = (const float*)d_in[6];
  const float* dec_w1  = (const float*)d_in[7];
  const float* dec_b1  = (const float*)d_in[8];
  const float* dec_w2  = (const float*)d_in[9];
  const float* dec_b2  = (const float*)d_in[10];
  const float* prior   = (const float*)d_in[11];
  float* out = (float*)d_out;

  float* ws = (float*)d_ws;
  float* h1    = ws;             // 32768
  float* h1g   = ws + 32768;     // 32768
  float* h     = ws + 65536;     // 32768
  float* z     = ws + 98304;     // 16384
  float* g     = ws + 114688;    // 32768
  float* x2    = ws + 147456;    // 256
  float* y2    = ws + 147712;    // 256
  float* uG    = ws + 147968;    // 256
  float* vG    = ws + 148224;    // 256
  float* recon = ws + 148480;    // 1
  float* Wacc  = ws + 148481;    // 1
  float* Cmat  = ws + 148736;    // 65536
  float* Kmat  = ws + 214272;    // 65536

  // 0) zero split-K accumulator + scalars
  init_kernel<<<dim3(128), dim3(256), 0, stream>>>(h1, recon, Wacc);
  // 1) encoder GEMM1 (split-K WMMA, async-LDS double-buffered)
  enc1_kernel<<<dim3(SPLITK, 16), dim3(256), 0, stream>>>(x, enc_w1, h1);
  // 2) bias + GELU
  biasgelu_kernel<<<dim3(128), dim3(256), 0, stream>>>(h1, enc_b1, h1g);
  // 3) encoder GEMM2 (+bias)
  small_gemm_kernel<H_SZ, H_SZ, false><<<dim3(16), dim3(256), 0, stream>>>(
      h1g, enc_w2, enc_b2, h);
  // 4) reparameterize
  z_kernel<<<dim3(64), dim3(256), 0, stream>>>(h, eps, z);
  // 5) decoder GEMM1 (+bias+GELU)
  small_gemm_kernel<DZ, DZ, true><<<dim3(16), dim3(256), 0, stream>>>(
      z, dec_w1, dec_b1, g);
  // 6) fused decoder GEMM2 + symlog recon loss (async-LDS staged weights)
  dec2_recon_kernel<<<dim3(512), dim3(256), 0, stream>>>(g, dec_w2, dec_b2, y,
                                                         recon);
  // 7) Sinkhorn setup
  rowsq_kernel<<<dim3(2), dim3(256), 0, stream>>>(z, prior, x2, y2);
  ck_kernel<<<dim3(32), dim3(256), 0, stream>>>(z, prior, x2, y2, Cmat, Kmat);
  // 8) Sinkhorn iterations (single workgroup)
  sinkhorn_iter_kernel<<<dim3(1), dim3(256), 0, stream>>>(Kmat, uG, vG);
  // 9) transport cost
  wsum_kernel<<<dim3(64), dim3(256), 0, stream>>>(Kmat, Cmat, uG, vG, Wacc);
  // 10) finalize
  finalize_kernel<<<dim3(1), dim3(1), 0, stream>>>(recon, Wacc, out);
}